// MCE_10943576670300
// MI455X (gfx1250) — compile-verified
//
#include <hip/hip_runtime.h>

#define N_USERS 100000
#define N_ITEMS 50000
#define N_NODES 150000
#define DD 64
#define EDGES 3000000

typedef __attribute__((ext_vector_type(8))) float v8f;
typedef __attribute__((ext_vector_type(2))) float v2f;

// ---------------------------------------------------------------------------
// Kernel 1: build ego = concat(pref, repre)  (float4 vectorized copy)
// ---------------------------------------------------------------------------
__global__ void mmgcn_init_ego(float* __restrict__ cur,
                               const float* __restrict__ pref,
                               const float* __restrict__ repre) {
    int i = blockIdx.x * blockDim.x + threadIdx.x;          // over N_NODES*64/4
    const int total = (N_NODES * DD) / 4;
    if (i >= total) return;
    const int userQ = (N_USERS * DD) / 4;
    float4 v = (i < userQ) ? ((const float4*)pref)[i]
                           : ((const float4*)repre)[i - userQ];
    ((float4*)cur)[i] = v;
}

// ---------------------------------------------------------------------------
// Kernel 2: edge scatter.  One wave per edge; lane l handles float2 at 2*l.
// Gather of cur[src] is 256B contiguous per wave; accumulation uses the
// native non-returning f32 atomic (global_atomic_add_f32, STOREcnt path) via
// inline asm so it can never fall back to a CAS loop. Both ego matrices are
// 38.4MB and L2-resident (192MB L2), so atomics resolve in L2.
// ---------------------------------------------------------------------------
__device__ __forceinline__ void atomic_add_f32_pair(float* addr, float v0, float v1) {
    unsigned long long a = (unsigned long long)(uintptr_t)addr;
    asm volatile("global_atomic_add_f32 %0, %1, off"          :: "v"(a), "v"(v0) : "memory");
    asm volatile("global_atomic_add_f32 %0, %1, off offset:4" :: "v"(a), "v"(v1) : "memory");
}

__global__ void mmgcn_scatter(const int*   __restrict__ ei,   // [2,E]
                              const float* __restrict__ ew,   // [E]
                              const float* __restrict__ x,    // cur ego
                              float*       __restrict__ acc)  // side (zeroed)
{
    int tid = blockIdx.x * blockDim.x + threadIdx.x;
    int e   = tid >> 5;
    if (e >= EDGES) return;                                   // wave-uniform
    int l   = tid & 31;
    int src = ei[e];
    int dst = ei[EDGES + e];
    float w = ew[e];
    float2 v = ((const float2*)(x + ((long long)src << 6)))[l];
    float* o = acc + ((long long)dst << 6) + 2 * l;
    atomic_add_f32_pair(o, v.x * w, v.y * w);
}

// ---------------------------------------------------------------------------
// Kernel 3: new_ego = side + 0.5*ego via V_WMMA_F32_16X16X4_F32.
//   D = sum_c I[:,4c:4c+4] x side[4c:4c+4, :] + 0.5*ego      (exact in f32)
// One wave per 16x16 tile; 4 chained WMMAs. Layout per ISA 7.12.2:
//   C/D: vgpr v, lane = (M = v + 8*(lane/16), N = lane%16)
//   A 16x4: lane%16 = M; lanes 0-15 hold K={0,1}, lanes 16-31 K={2,3}
//   B 4x16: lane%16 = N; lanes 0-15 hold K={0,1}, lanes 16-31 K={2,3}
// ---------------------------------------------------------------------------
__global__ void mmgcn_axpy_wmma(const float* __restrict__ side,
                                const float* __restrict__ cur,
                                float*       __restrict__ outp,
                                int numTiles)
{
    int wave = (blockIdx.x * blockDim.x + threadIdx.x) >> 5;
    if (wave >= numTiles) return;                             // wave-uniform
    int lane = threadIdx.x & 31;
    int half = lane >> 4;
    int l16  = lane & 15;
    int tr = wave >> 2;                                       // 64/16 = 4 col tiles
    int tc = wave & 3;
    int r0 = tr << 4, c0 = tc << 4;
    const int ld = DD;

    v8f acc;
#pragma unroll
    for (int v = 0; v < 8; ++v)
        acc[v] = 0.5f * cur[(r0 + v + 8 * half) * ld + c0 + l16];

#pragma unroll
    for (int c = 0; c < 4; ++c) {
        v2f a, b;
#pragma unroll
        for (int v = 0; v < 2; ++v) {
            int kl = 2 * half + v;                            // K within chunk
            a[v] = (l16 == 4 * c + kl) ? 1.0f : 0.0f;         // identity cols
            b[v] = side[(r0 + 4 * c + kl) * ld + c0 + l16];   // side rows
        }
        acc = __builtin_amdgcn_wmma_f32_16x16x4_f32(
                  false, a, false, b, (short)0, acc, false, false);
    }

#pragma unroll
    for (int v = 0; v < 8; ++v)
        outp[(r0 + v + 8 * half) * ld + c0 + l16] = acc[v];
}

// ---------------------------------------------------------------------------
// Kernel 4: write final ego of modality m into column-interleaved output.
//   users:  out[n*128 + m*64 + d],  items: out[USER_OFF + (n-NU)*128 + m*64 + d]
// ---------------------------------------------------------------------------
__global__ void mmgcn_writeout(const float* __restrict__ ego,
                               float* __restrict__ out, int mod) {
    int i = blockIdx.x * blockDim.x + threadIdx.x;            // over N_NODES*16
    const int total = N_NODES * (DD / 4);
    if (i >= total) return;
    int n  = i >> 4;
    int dq = i & 15;
    float4 v = ((const float4*)ego)[i];
    long long off;
    if (n < N_USERS)
        off = (long long)n * 128 + mod * 64 + dq * 4;
    else
        off = (long long)N_USERS * 128 + (long long)(n - N_USERS) * 128
            + mod * 64 + dq * 4;
    *(float4*)(out + off) = v;
}

// ---------------------------------------------------------------------------
extern "C" void kernel_launch(void* const* d_in, const int* in_sizes, int n_in,
                              void* d_out, int out_size, void* d_ws, size_t ws_size,
                              hipStream_t stream) {
    const size_t matElems = (size_t)N_NODES * DD;             // 9.6M floats
    float* bufA = (float*)d_ws;
    float* bufB = bufA + matElems;

    const int initBlocks    = (int)((matElems / 4 + 255) / 256);          // 9375
    const int scatterBlocks = (int)(((long long)EDGES * 32 + 255) / 256); // 375000
    const int numTiles      = (N_NODES / 16) * (DD / 16);                 // 37500
    const int axpyBlocks    = (numTiles + 7) / 8;                         // 8 waves/block
    const int woBlocks      = (int)((matElems / 4 + 255) / 256);

    for (int m = 0; m < 2; ++m) {
        const int*   ei    = (const int*)  d_in[m ? 2 : 0];
        const float* ew    = (const float*)d_in[m ? 3 : 1];
        const float* pref  = (const float*)d_in[m ? 5 : 4];
        const float* repre = (const float*)d_in[m ? 7 : 6];

        float* cur  = bufA;
        float* side = bufB;

        mmgcn_init_ego<<<initBlocks, 256, 0, stream>>>(cur, pref, repre);

        for (int layer = 0; layer < 3; ++layer) {
            hipMemsetAsync(side, 0, matElems * sizeof(float), stream);
            mmgcn_scatter<<<scatterBlocks, 256, 0, stream>>>(ei, ew, cur, side);
            // new ego written in place into `side`, then swap
            mmgcn_axpy_wmma<<<axpyBlocks, 256, 0, stream>>>(side, cur, side, numTiles);
            float* t = cur; cur = side; side = t;
        }

        mmgcn_writeout<<<woBlocks, 256, 0, stream>>>(cur, (float*)d_out, m);
    }
}